// SupConLoss_44066364457578
// MI455X (gfx1250) — compile-verified
//
#include <hip/hip_runtime.h>
#include <math.h>

// SupCon loss, fused flash-style, fp32 WMMA (v_wmma_f32_16x16x4_f32) on gfx1250.
// B=32 batches, N=2048 bank, D=128 feature dim (fixed by the reference).

typedef float v2f __attribute__((ext_vector_type(2)));
typedef float v8f __attribute__((ext_vector_type(8)));
typedef int   v4i __attribute__((ext_vector_type(4)));

#define BB 32
#define NN 2048
#define DD 128
#define INVT (1.0f / 0.07f)

#define AS1 __attribute__((address_space(1)))
#define AS3 __attribute__((address_space(3)))
typedef AS1 v4i* gp_t;   // global int4*
typedef AS3 v4i* lp_t;   // LDS int4*

#if defined(__gfx1250__) && __has_builtin(__builtin_amdgcn_global_load_async_to_lds_b128)
#define USE_ASYNC_LDS 1
#else
#define USE_ASYNC_LDS 0
#endif

// ---------------- Pass 1: L2 normalize rows + labels int64->int32 ----------------
__global__ __launch_bounds__(256) void supcon_normalize(
    const float* __restrict__ feat, const long long* __restrict__ labels,
    float* __restrict__ fn, int* __restrict__ lab32)
{
    const int wave = threadIdx.x >> 5;
    const int lane = threadIdx.x & 31;
    const int row  = blockIdx.x * 8 + wave;            // B*N = 65536 rows total
    const float4* src = (const float4*)(feat + (size_t)row * DD);
    float4 v = src[lane];                              // 4 floats/lane * 32 lanes = 128
    float ss = v.x * v.x + v.y * v.y + v.z * v.z + v.w * v.w;
#pragma unroll
    for (int o = 16; o > 0; o >>= 1) ss += __shfl_xor(ss, o, 32);
    float r = 1.0f / fmaxf(sqrtf(ss), 1e-12f);         // torch F.normalize eps
    float4 o4; o4.x = v.x * r; o4.y = v.y * r; o4.z = v.z * r; o4.w = v.w * r;
    ((float4*)(fn + (size_t)row * DD))[lane] = o4;
    if (lane == 0) lab32[row] = (int)labels[row];
}

// ---------------- Pass 2: fused Gram + online softmax + masked stats ----------------
// Block = 128 threads (4 waves). Wave w owns rows [itile, itile+16).
// Block double-buffers 16x128 j-tiles in LDS (async DMA when available);
// each wave runs 32 fp32 WMMAs per tile.
__global__ __launch_bounds__(128) void supcon_main(
    const float* __restrict__ fn, const int* __restrict__ lab32,
    float* __restrict__ loss)
{
    __shared__ float tile[2][16 * DD];   // 2 x 8KB double buffer
    __shared__ int   jlabAll[NN];        // 8KB: all labels of this batch

    const int b     = blockIdx.y;
    const int wv    = threadIdx.x >> 5;
    const int lane  = threadIdx.x & 31;
    const int t     = threadIdx.x;
    const int itile = blockIdx.x * 64 + wv * 16;
    const float* Fb = fn + (size_t)b * NN * DD;
    const int*   Lb = lab32 + b * NN;

    // A-matrix fragment layout for V_WMMA_F32_16X16X4_F32 (ISA 7.12.2):
    // lane l: M = l&15 ; VGPR pair holds K = {2*(l>>4), 2*(l>>4)+1} within each K-step.
    const int rsel = lane & 15;
    const int ksel = (lane >> 4) * 2;

    v2f areg[32];
#pragma unroll
    for (int kk = 0; kk < 32; ++kk)
        areg[kk] = *(const v2f*)(Fb + (size_t)(itile + rsel) * DD + kk * 4 + ksel);

    int labI[8];
#pragma unroll
    for (int r = 0; r < 8; ++r)
        labI[r] = Lb[itile + r + 8 * (lane >> 4)];

    // Online softmax state per (lane, acc-slot): row = itile + r + 8*(lane>>4)
    float m[8], den[8], sp[8], np[8];
#pragma unroll
    for (int r = 0; r < 8; ++r) { m[r] = -INFINITY; den[r] = 0.f; sp[r] = 0.f; np[r] = 0.f; }

    // Preload all batch labels into LDS (once)
    for (int i = t; i < NN; i += 128) jlabAll[i] = Lb[i];

#if USE_ASYNC_LDS
    // ---- async Global->LDS staging (ASYNCcnt-tracked DMA, overlaps WMMA) ----
    // Tile = 8KB; thread t owns slots {t + 128q}: same 2048-byte stride on both the
    // global and LDS side, so one address pair + 4 immediate offsets (ISA: INST_OFFSET
    // is applied to both the VADDR and VDST addresses).
#define STAGE_TILE(JT, BUF)                                                          \
    {                                                                                \
        gp_t gsrc = (gp_t)(Fb + (size_t)(JT) * 16 * DD + t * 4);                     \
        lp_t ldst = (lp_t)&tile[BUF][t * 4];                                         \
        __builtin_amdgcn_global_load_async_to_lds_b128(gsrc, ldst, 0, 0);            \
        __builtin_amdgcn_global_load_async_to_lds_b128(gsrc, ldst, 2048, 0);         \
        __builtin_amdgcn_global_load_async_to_lds_b128(gsrc, ldst, 4096, 0);         \
        __builtin_amdgcn_global_load_async_to_lds_b128(gsrc, ldst, 6144, 0);         \
    }
#if __has_builtin(__builtin_amdgcn_s_wait_asynccnt)
#define WAIT_ASYNC0() __builtin_amdgcn_s_wait_asynccnt(0)
#else
#define WAIT_ASYNC0() asm volatile("s_wait_asynccnt 0x0" ::: "memory")
#endif

    STAGE_TILE(0, 0);
    WAIT_ASYNC0();
    __syncthreads();

    const int NT = NN / 16;  // 128 j-tiles
    for (int jt = 0; jt < NT; ++jt) {
        const int cur = jt & 1;
        if (jt + 1 < NT) STAGE_TILE(jt + 1, cur ^ 1);   // fire-and-forget DMA
#else
    // ---- fallback: register staging ----
    float4 sreg[4];
    {
        const float4* src = (const float4*)(Fb);
#pragma unroll
        for (int q = 0; q < 4; ++q) sreg[q] = src[t + 128 * q];
        float4* dst = (float4*)&tile[0][0];
#pragma unroll
        for (int q = 0; q < 4; ++q) dst[t + 128 * q] = sreg[q];
    }
    __syncthreads();

    const int NT = NN / 16;  // 128 j-tiles
    for (int jt = 0; jt < NT; ++jt) {
        const int cur = jt & 1;
        if (jt + 1 < NT) {
            const float4* src = (const float4*)(Fb + (size_t)(jt + 1) * 16 * DD);
#pragma unroll
            for (int q = 0; q < 4; ++q) sreg[q] = src[t + 128 * q];
        }
#endif

        // 16x16x128 = 32 chained v_wmma_f32_16x16x4_f32, B-frags from LDS
        v8f acc = {0.f, 0.f, 0.f, 0.f, 0.f, 0.f, 0.f, 0.f};
        const float* tb = &tile[cur][0];
#pragma unroll
        for (int kk = 0; kk < 32; ++kk) {
            v2f bfrag = *(const v2f*)(tb + rsel * DD + kk * 4 + ksel);
            acc = __builtin_amdgcn_wmma_f32_16x16x4_f32(
                false, areg[kk], false, bfrag, (short)0, acc, false, false);
        }

        // Epilogue: C layout => VGPR r, lane l: row = r + 8*(l>>4), col = l&15
        const int j    = jt * 16 + rsel;
        const int labj = jlabAll[j];
#pragma unroll
        for (int r = 0; r < 8; ++r) {
            float s = acc[r] * INVT;
            const int  rowg = itile + r + 8 * (lane >> 4);
            const bool self = (j == rowg);
            float nm = fmaxf(m[r], s);
            if (nm > m[r]) { den[r] *= __expf(m[r] - nm); m[r] = nm; }  // rare rescale
            if (!self) {
                den[r] += __expf(s - m[r]);
                if (labj == labI[r]) { sp[r] += s; np[r] += 1.f; }
            }
        }

#if USE_ASYNC_LDS
        WAIT_ASYNC0();           // DMA into the other buffer has landed
#else
        if (jt + 1 < NT) {
            float4* dst = (float4*)&tile[cur ^ 1][0];
#pragma unroll
            for (int q = 0; q < 4; ++q) dst[t + 128 * q] = sreg[q];
        }
#endif
        __syncthreads();
    }

    // Merge 16 column-lanes per row (xor masks < 16 stay inside each half-wave)
#pragma unroll
    for (int r = 0; r < 8; ++r) {
        float mm = m[r], dd = den[r], ss = sp[r], nn = np[r];
#pragma unroll
        for (int o = 1; o < 16; o <<= 1) {
            float mo  = __shfl_xor(mm, o, 32);
            float do2 = __shfl_xor(dd, o, 32);
            float so  = __shfl_xor(ss, o, 32);
            float no  = __shfl_xor(nn, o, 32);
            float nm2 = fmaxf(mm, mo);
            dd = dd * __expf(mm - nm2) + do2 * __expf(mo - nm2);
            mm = nm2; ss += so; nn += no;
        }
        if ((lane & 15) == 0) {
            const int rowg = itile + r + 8 * (lane >> 4);
            loss[b * NN + rowg] = logf(dd) + mm - ss / nn;
        }
    }
}

// ---------------- Pass 3: loss_m = sum(loss) / count_nonzero(loss) ----------------
__global__ __launch_bounds__(256) void supcon_mean(
    const float* __restrict__ loss, float* __restrict__ out0)
{
    __shared__ float ssum[256];
    __shared__ float scnt[256];
    float s = 0.f, c = 0.f;
    for (int i = threadIdx.x; i < BB * NN; i += 256) {
        float v = loss[i];
        s += v;
        if (v != 0.f) c += 1.f;
    }
    ssum[threadIdx.x] = s; scnt[threadIdx.x] = c;
    __syncthreads();
    for (int o = 128; o > 0; o >>= 1) {
        if (threadIdx.x < o) {
            ssum[threadIdx.x] += ssum[threadIdx.x + o];
            scnt[threadIdx.x] += scnt[threadIdx.x + o];
        }
        __syncthreads();
    }
    if (threadIdx.x == 0) out0[0] = ssum[0] / scnt[0];
}

extern "C" void kernel_launch(void* const* d_in, const int* in_sizes, int n_in,
                              void* d_out, int out_size, void* d_ws, size_t ws_size,
                              hipStream_t stream) {
    const float*     feat   = (const float*)d_in[0];      // [32,2048,128] f32
    const long long* labels = (const long long*)d_in[1];  // [32,2048] i64
    float* out = (float*)d_out;                           // [0]=loss_m, [1..]=loss

    float* fn    = (float*)d_ws;                          // 32*2048*128 f32 = 32 MB
    int*   lab32 = (int*)((char*)d_ws + (size_t)BB * NN * DD * sizeof(float));

    // Pass 1: 65536 rows, 8 rows per 256-thread block
    supcon_normalize<<<(BB * NN) / 8, 256, 0, stream>>>(feat, labels, fn, lab32);

    // Pass 2: grid = (N/64 strips, B batches), 128 threads (4 waves, 16 rows each)
    dim3 grid(NN / 64, BB);
    supcon_main<<<grid, 128, 0, stream>>>(fn, lab32, out + 1);

    // Pass 3: single block reduction
    supcon_mean<<<1, 256, 0, stream>>>(out + 1, out);
}